// DependencyMemory_49581102465078
// MI455X (gfx1250) — compile-verified
//
#include <hip/hip_runtime.h>

typedef float v2f __attribute__((ext_vector_type(2)));
typedef float v8f __attribute__((ext_vector_type(8)));

#define NB 16
#define NS 512
#define NM 16
#define ND 1024
#define BDIM 256
#define ESTRIDE 1028   // padded LDS row stride (floats): 1028 % 64 == 4 -> conflict-free
                       // WMMA A feed; 1028 % 4 == 0 -> 16B-aligned rows for float4

__global__ __launch_bounds__(BDIM) void dep_mem_attn_kernel(
    const float* __restrict__ q_g,        // [B,S,D]
    const int*   __restrict__ dep_text,   // [B,S,M]
    const int*   __restrict__ dep_type,   // [B,S,M]
    const int*   __restrict__ mem_mask,   // [B,S,M]
    const float* __restrict__ out_mask,   // [B,S]
    const float* __restrict__ text_tab,   // [32000,D]
    const float* __restrict__ type_tab,   // [64,D]
    float*       __restrict__ out)        // [B,S,D]
{
    __shared__ float s_embed[NM * ESTRIDE];   // 65792 B
    __shared__ float s_q[ND];                 //  4096 B
    __shared__ float s_part[8][16];
    __shared__ float s_scores[16];
    __shared__ float s_probs[16];
    __shared__ int   s_idx[2][16];

    const int row  = blockIdx.x;              // b*S + s
    const int tid  = threadIdx.x;
    const int lane = tid & 31;
    const int wave = tid >> 5;

    // ---- stage 0: indices + query into LDS -------------------------------
    if (tid < NM) {
        s_idx[0][tid] = dep_text[(size_t)row * NM + tid];
        s_idx[1][tid] = dep_type[(size_t)row * NM + tid];
    }
    {
        const float4* qp = (const float4*)(q_g + (size_t)row * ND);
        ((float4*)s_q)[tid] = qp[tid];        // 256 threads x float4 = 1024 floats
    }
    __syncthreads();

    // ---- stage 1: coalesced gather of 16 embedding rows into LDS ---------
    // per m: 256 threads each load one float4 from text/type tables (L2-resident)
    for (int m = 0; m < NM; ++m) {
        const float4* tp = (const float4*)(text_tab + (size_t)s_idx[0][m] * ND);
        const float4* yp = (const float4*)(type_tab + (size_t)s_idx[1][m] * ND);
        float4 a = tp[tid];
        float4 b = yp[tid];
        float4 e; e.x = a.x + b.x; e.y = a.y + b.y; e.z = a.z + b.z; e.w = a.w + b.w;
        *(float4*)(s_embed + m * ESTRIDE + tid * 4) = e;   // 16B-aligned
    }
    __syncthreads();

    // ---- stage 2: scores via V_WMMA_F32_16X16X4_F32 ----------------------
    // A = embed tile (16xK), B = q replicated into 16 columns -> every
    // column of C is the score vector. 8 waves each own a K=128 chunk.
    const int   m_l  = lane & 15;
    const int   hi2  = (lane >> 4) << 1;                 // 0 or 2
    const float* arow = s_embed + m_l * ESTRIDE + hi2;
    const float* brow = s_q + hi2;
    const int   kbase = wave * (ND / 8);                 // 128 per wave

    v8f c = {};
    #pragma unroll 8
    for (int k = 0; k < ND / 8; k += 4) {
        v2f a = *(const v2f*)(arow + kbase + k);         // A: (M=m_l, K=k+hi2+{0,1})
        v2f b = *(const v2f*)(brow + kbase + k);         // B: all N cols = q
        c = __builtin_amdgcn_wmma_f32_16x16x4_f32(
                /*neg_a=*/false, a, /*neg_b=*/false, b,
                /*c_mod=*/(short)0, c, /*reuse_a=*/false, /*reuse_b=*/false);
    }

    // C layout: VGPR r, lanes 0-15 -> M=r (N=lane), lanes 16-31 -> M=8+r.
    if (lane == 0 || lane == 16) {
        const int mb = (lane >> 4) * 8;
        #pragma unroll
        for (int r = 0; r < 8; ++r) s_part[wave][mb + r] = c[r];
    }
    __syncthreads();

    // ---- stage 3: reduce partials, mask, softmax over M=16 ---------------
    if (tid < NM) {
        float sc = 0.0f;
        #pragma unroll
        for (int w = 0; w < 8; ++w) sc += s_part[w][tid];
        sc *= (1.0f / 32.0f);                             // / sqrt(1024)
        if (mem_mask[(size_t)row * NM + tid] == 0) sc = -1e9f;
        s_scores[tid] = sc;
    }
    __syncthreads();
    if (tid == 0) {
        float mx = s_scores[0];
        #pragma unroll
        for (int i = 1; i < NM; ++i) mx = fmaxf(mx, s_scores[i]);
        float e[NM];
        float sum = 0.0f;
        #pragma unroll
        for (int i = 0; i < NM; ++i) { e[i] = __expf(s_scores[i] - mx); sum += e[i]; }
        const float inv = 1.0f / sum;
        #pragma unroll
        for (int i = 0; i < NM; ++i) s_probs[i] = e[i] * inv;
    }
    __syncthreads();

    // ---- stage 4: out = (probs . embed) * out_mask + q -------------------
    float p[NM];
    #pragma unroll
    for (int i = 0; i < NM; ++i) p[i] = s_probs[i];
    const float om = out_mask[row];

    float4 acc; acc.x = acc.y = acc.z = acc.w = 0.0f;
    const float* base = s_embed + tid * 4;
    #pragma unroll
    for (int m = 0; m < NM; ++m) {
        float4 ev = *(const float4*)(base + m * ESTRIDE);
        acc.x += p[m] * ev.x; acc.y += p[m] * ev.y;
        acc.z += p[m] * ev.z; acc.w += p[m] * ev.w;
    }
    float4 qv = ((const float4*)s_q)[tid];
    float4 o;
    o.x = acc.x * om + qv.x; o.y = acc.y * om + qv.y;
    o.z = acc.z * om + qv.z; o.w = acc.w * om + qv.w;
    ((float4*)(out + (size_t)row * ND))[tid] = o;
}

extern "C" void kernel_launch(void* const* d_in, const int* in_sizes, int n_in,
                              void* d_out, int out_size, void* d_ws, size_t ws_size,
                              hipStream_t stream) {
    (void)in_sizes; (void)n_in; (void)out_size; (void)d_ws; (void)ws_size;
    const float* q  = (const float*)d_in[0];
    const int*   dt = (const int*)  d_in[1];
    const int*   dy = (const int*)  d_in[2];
    const int*   mm = (const int*)  d_in[3];
    const float* om = (const float*)d_in[4];
    const float* tt = (const float*)d_in[5];
    const float* yt = (const float*)d_in[6];
    float* outp = (float*)d_out;

    dim3 grid(NB * NS);   // 8192 workgroups, one per (b,s)
    dim3 block(BDIM);     // 8 waves (wave32)
    dep_mem_attn_kernel<<<grid, block, 0, stream>>>(q, dt, dy, mm, om, tt, yt, outp);
}